// DynamicOffsetPatcher_81801947119831
// MI455X (gfx1250) — compile-verified
//
#include <hip/hip_runtime.h>
#include <hip/hip_bf16.h>

typedef float v2f __attribute__((ext_vector_type(2)));
typedef float v8f __attribute__((ext_vector_type(8)));

#define B_  64
#define L_  512
#define N_  64
#define A_  64
#define D_  256
#define P_  16

// ---------------------------------------------------------------------------
// Kernel 1: Conv1d(N->32, k=3, pad=1) + bias + ReLU, mean-pool over L.
// One block per batch b; 256 threads = (8 l-groups) x (32 channels).
// x chunked through LDS; conv weights padded to pitch 194 (194 % 64 = 2)
// so the 32 lanes of a wave (distinct c) hit distinct LDS banks.
// ---------------------------------------------------------------------------
__global__ __launch_bounds__(256) void k_conv_pool(
    const float* __restrict__ x,       // (B, L, N)
    const float* __restrict__ conv_w,  // (32, N, 3)
    const float* __restrict__ conv_b,  // (32,)
    float* __restrict__ pooled)        // (B, 32)
{
    __shared__ float w_s[32 * 194];
    __shared__ float x_s[66 * 64];
    __shared__ float red[256];

    const int b = blockIdx.x;
    const int t = threadIdx.x;

    for (int i = t; i < 32 * 192; i += 256) {
        int c = i / 192, r = i % 192;
        w_s[c * 194 + r] = conv_w[i];
    }

    const int c  = t & 31;
    const int lg = t >> 5;
    const float bias = conv_b[c];
    const float* xb = x + (size_t)b * (L_ * N_);
    float acc = 0.0f;

    for (int l0 = 0; l0 < L_; l0 += 64) {
        __syncthreads();
        for (int i = t; i < 66 * 64; i += 256) {
            int gl = l0 - 1 + (i >> 6);
            int n  = i & 63;
            x_s[i] = (gl >= 0 && gl < L_) ? xb[gl * 64 + n] : 0.0f;
        }
        __syncthreads();
        const float* wc = &w_s[c * 194];
        for (int l = lg; l < 64; l += 8) {
            // output position l0+l taps local rows l, l+1, l+2
            const float* x0 = &x_s[(l + 0) * 64];
            const float* x1 = &x_s[(l + 1) * 64];
            const float* x2 = &x_s[(l + 2) * 64];
            float s = bias;
            #pragma unroll 8
            for (int n = 0; n < 64; ++n) {
                s = fmaf(x0[n], wc[n * 3 + 0], s);
                s = fmaf(x1[n], wc[n * 3 + 1], s);
                s = fmaf(x2[n], wc[n * 3 + 2], s);
            }
            acc += fmaxf(s, 0.0f);
        }
    }

    red[t] = acc;
    __syncthreads();
    if (t < 32) {
        float s = 0.0f;
        #pragma unroll
        for (int g = 0; g < 8; ++g) s += red[g * 32 + t];
        pooled[b * 32 + t] = s * (1.0f / (float)L_);
    }
}

// ---------------------------------------------------------------------------
// Kernel 2: delta[b,a] = 4 * tanh( pooled[b,:] . lin_w[a,:] + lin_b[a] )
// ---------------------------------------------------------------------------
__global__ __launch_bounds__(256) void k_delta(
    const float* __restrict__ pooled,  // (B, 32)
    const float* __restrict__ lin_w,   // (A, 32)
    const float* __restrict__ lin_b,   // (A,)
    float* __restrict__ delta)         // (B, A)
{
    int idx = blockIdx.x * 256 + threadIdx.x;
    if (idx >= B_ * A_) return;
    int b = idx >> 6, a = idx & 63;
    float s = lin_b[a];
    #pragma unroll 8
    for (int ci = 0; ci < 32; ++ci)
        s = fmaf(pooled[b * 32 + ci], lin_w[a * 32 + ci], s);
    delta[idx] = 4.0f * tanhf(s);
}

// ---------------------------------------------------------------------------
// Kernel 3: bilinear gather + patch projection via V_WMMA_F32_16X16X4_F32.
// Block = one (b,a): out[b, :, a, :] = samp(64x16) @ wp_w^T(16x256) + wp_b.
// 8 waves: wave w -> n-tile (w&3), d-tile group (w>>2)*8 .. +7.
// A-frag: lane m = series n (coalesced gathers); K striped per ISA layout.
// B-frag from wp_w^T staged in LDS (pitch 260 -> conflict-free).
// ---------------------------------------------------------------------------
__global__ __launch_bounds__(256) void k_main(
    const float* __restrict__ x,       // (B, L, N)
    const float* __restrict__ wp_w,    // (D, P)
    const float* __restrict__ wp_b,    // (D,)
    const float* __restrict__ delta,   // (B, A)
    float* __restrict__ out)           // (B*N, A, D)
{
    __shared__ float wp_s[P_ * 260];   // wp^T[p][d], pitch 260
    __shared__ float wb_s[D_];
    __shared__ int   i0_s[P_];
    __shared__ int   i1_s[P_];
    __shared__ float f_s[P_];

    const int blk = blockIdx.x;
    const int b = blk >> 6;
    const int a = blk & 63;
    const int t = threadIdx.x;

    for (int i = t; i < D_ * P_; i += 256) {
        int d = i >> 4, p = i & 15;
        wp_s[p * 260 + d] = wp_w[i];
    }
    wb_s[t] = wp_b[t];
    if (t < P_) {
        float xs = (float)(a * 8) + delta[b * A_ + a] + (float)t - 7.5f;
        xs = fminf(fmaxf(xs, 0.0f), (float)(L_ - 1));
        int i0 = (int)floorf(xs);
        if (i0 > L_ - 1) i0 = L_ - 1;
        float f = xs - (float)i0;
        int i1 = i0 + 1; if (i1 > L_ - 1) i1 = L_ - 1;
        i0_s[t] = i0; i1_s[t] = i1; f_s[t] = f;
    }
    __syncthreads();

    const int w      = t >> 5;
    const int lane   = t & 31;
    const int half   = lane >> 4;      // lane-half selects K pair / output row group
    const int m      = lane & 15;
    const int ntile  = w & 3;
    const int dgroup = w >> 2;
    const int n      = ntile * 16 + m;
    const float* xb  = x + (size_t)b * (L_ * N_) + n;

    // A fragments: samp[n][p] gathered+lerped once, reused over 8 d-tiles.
    v2f aF[4];
    #pragma unroll
    for (int kk = 0; kk < 4; ++kk) {
        int p0 = kk * 4 + half * 2;
        #pragma unroll
        for (int v = 0; v < 2; ++v) {
            int p   = p0 + v;
            float g0 = xb[i0_s[p] * N_];
            float g1 = xb[i1_s[p] * N_];
            float f  = f_s[p];
            aF[kk][v] = g0 * (1.0f - f) + g1 * f;
        }
    }

    for (int j = 0; j < 8; ++j) {
        int dt   = dgroup * 8 + j;
        int dcol = dt * 16 + m;
        v8f acc = {};
        #pragma unroll
        for (int kk = 0; kk < 4; ++kk) {
            int p0 = kk * 4 + half * 2;
            v2f bF;
            bF[0] = wp_s[(p0 + 0) * 260 + dcol];
            bF[1] = wp_s[(p0 + 1) * 260 + dcol];
            acc = __builtin_amdgcn_wmma_f32_16x16x4_f32(
                false, aF[kk], false, bF, (short)0, acc, false, false);
        }
        float wb = wb_s[dcol];
        #pragma unroll
        for (int r = 0; r < 8; ++r) {
            int nrow = ntile * 16 + half * 8 + r;       // C-frag M = half*8 + r
            out[((((size_t)b * N_ + nrow) * A_ + a) << 8) + dcol] = acc[r] + wb;
        }
    }
}

// ---------------------------------------------------------------------------
extern "C" void kernel_launch(void* const* d_in, const int* in_sizes, int n_in,
                              void* d_out, int out_size, void* d_ws, size_t ws_size,
                              hipStream_t stream) {
    const float* x      = (const float*)d_in[0];
    const float* conv_w = (const float*)d_in[1];
    const float* conv_b = (const float*)d_in[2];
    const float* lin_w  = (const float*)d_in[3];
    const float* lin_b  = (const float*)d_in[4];
    const float* wp_w   = (const float*)d_in[5];
    const float* wp_b   = (const float*)d_in[6];
    float* out = (float*)d_out;

    float* pooled = (float*)d_ws;          // B*32  = 2048 floats
    float* delta  = pooled + B_ * 32;      // B*A   = 4096 floats

    k_conv_pool<<<B_, 256, 0, stream>>>(x, conv_w, conv_b, pooled);
    k_delta<<<(B_ * A_ + 255) / 256, 256, 0, stream>>>(pooled, lin_w, lin_b, delta);
    k_main<<<B_ * A_, 256, 0, stream>>>(x, wp_w, wp_b, delta, out);
}